// PriorLoss_29515015258307
// MI455X (gfx1250) — compile-verified
//
#include <hip/hip_runtime.h>
#include <math.h>

#define V_SIZE 30000
#define C_B    16
#define S_LEN  1024
#define D_DIM  512
#define P_PAIRS 50000
#define EPSI   1e-8f

typedef __attribute__((ext_vector_type(2))) float v2f;
typedef __attribute__((ext_vector_type(8))) float v8f;

// ---------------- workspace layout ----------------
constexpr size_t alignup(size_t x) { return (x + 255) & ~(size_t)255; }
constexpr size_t OFF_ACC  = 0;                                      // 2 floats (numer, denom)
constexpr size_t OFF_USED = alignup(OFF_ACC + 2 * 4);               // V ints
constexpr size_t OFF_SRCF = alignup(OFF_USED + (size_t)V_SIZE * 4); // V ints
constexpr size_t OFF_TGTF = alignup(OFF_SRCF + (size_t)V_SIZE * 4); // V ints
constexpr size_t OFF_TF   = alignup(OFF_TGTF + (size_t)V_SIZE * 4); // C*S ints
constexpr size_t OFF_SRCA = alignup(OFF_TF   + (size_t)C_B * S_LEN * 4);
constexpr size_t OFF_TGTA = alignup(OFF_SRCA + (size_t)C_B * S_LEN * 4);
constexpr size_t OFF_POS  = alignup(OFF_TGTA + (size_t)C_B * S_LEN * 4); // C*V ints
constexpr size_t OFF_BITS = alignup(OFF_POS  + (size_t)C_B * V_SIZE * 4); // C*S*(S/32) u32
constexpr size_t OFF_UEFF = alignup(OFF_BITS + (size_t)C_B * S_LEN * (S_LEN / 32) * 4); // C*S*D f32
constexpr size_t OFF_VEFF = alignup(OFF_UEFF + (size_t)C_B * S_LEN * D_DIM * 4);        // C*S*D f32
constexpr size_t WS_NEED_ASYNC = OFF_VEFF + (size_t)C_B * S_LEN * D_DIM * 4;

// ---------------- prep kernels ----------------
__global__ void k_fill(int* __restrict__ p, int val, int n) {
    for (int i = blockIdx.x * blockDim.x + threadIdx.x; i < n; i += gridDim.x * blockDim.x)
        p[i] = val;
}

// tokens -> used flags, tf_mask, per-batch min position of each token
__global__ void k_scan(const int* __restrict__ gt, const unsigned char* __restrict__ btg,
                       const unsigned char* __restrict__ btf,
                       int* __restrict__ used, int* __restrict__ tf_mask, int* __restrict__ pos) {
    int idx = blockIdx.x * blockDim.x + threadIdx.x;
    if (idx >= C_B * S_LEN) return;
    int token = gt[idx];
    int tf = (btf[idx] != 0) && (btg[idx] != 0);
    tf_mask[idx] = tf;
    atomicOr(&used[token], 1);
    int b = idx >> 10, s = idx & (S_LEN - 1);
    atomicMin(&pos[(size_t)b * V_SIZE + token], s);
}

// kept pairs -> src/tgt token flags
__global__ void k_pairs(const int* __restrict__ src, const int* __restrict__ tgt,
                        const int* __restrict__ used,
                        int* __restrict__ srcf, int* __restrict__ tgtf) {
    int p = blockIdx.x * blockDim.x + threadIdx.x;
    if (p >= P_PAIRS) return;
    int s = src[p], t = tgt[p];
    if (used[s] && used[t]) { atomicOr(&srcf[s], 1); atomicOr(&tgtf[t], 1); }
}

__global__ void k_assoc(const int* __restrict__ gt, const int* __restrict__ tf_mask,
                        const int* __restrict__ srcf, const int* __restrict__ tgtf,
                        int* __restrict__ sa, int* __restrict__ ta) {
    int idx = blockIdx.x * blockDim.x + threadIdx.x;
    if (idx >= C_B * S_LEN) return;
    int token = gt[idx];
    sa[idx] = (srcf[token] && tf_mask[idx]) ? 1 : 0;
    ta[idx] = tgtf[token] ? 1 : 0;
}

// sparse target matrix as bit-mask: target[b][sp][tp] = 1
__global__ void k_target(const int* __restrict__ src, const int* __restrict__ tgt,
                         const int* __restrict__ used, const int* __restrict__ pos,
                         const int* __restrict__ tf_mask, unsigned* __restrict__ bits) {
    int idx = blockIdx.x * blockDim.x + threadIdx.x;
    if (idx >= C_B * P_PAIRS) return;
    int b = idx / P_PAIRS, p = idx - b * P_PAIRS;
    int s = src[p], t = tgt[p];
    if (!(used[s] && used[t])) return;
    int sp = pos[(size_t)b * V_SIZE + s];
    int tp = pos[(size_t)b * V_SIZE + t];
    if (sp >= S_LEN || tp >= S_LEN) return;               // token absent in this batch row
    if (!tf_mask[b * S_LEN + sp]) return;                 // src_tf_valid
    atomicOr(&bits[((size_t)(b * S_LEN + sp) << 5) + (tp >> 5)], 1u << (tp & 31));
}

// materialize effective factors once: u_eff = u*u_score*tf, v_eff = v*v_score
__global__ void k_eff(const float* __restrict__ u,  const float* __restrict__ us,
                      const float* __restrict__ v,  const float* __restrict__ vs,
                      const int* __restrict__ tfm,
                      float* __restrict__ ueff, float* __restrict__ veff) {
    int i = blockIdx.x * blockDim.x + threadIdx.x;        // float4 index
    const int total = C_B * S_LEN * D_DIM / 4;
    if (i >= total) return;
    float4 a = ((const float4*)u)[i];
    float4 s = ((const float4*)us)[i];
    float tf = tfm[i / (D_DIM / 4)] ? 1.f : 0.f;
    float4 oa; oa.x = a.x * s.x * tf; oa.y = a.y * s.y * tf;
               oa.z = a.z * s.z * tf; oa.w = a.w * s.w * tf;
    ((float4*)ueff)[i] = oa;
    float4 b = ((const float4*)v)[i];
    float4 t = ((const float4*)vs)[i];
    float4 ob; ob.x = b.x * t.x; ob.y = b.y * t.y;
               ob.z = b.z * t.z; ob.w = b.w * t.w;
    ((float4*)veff)[i] = ob;
}

// ---------------- shared epilogue: fused BCE loss + reduction ----------------
__device__ __forceinline__ void loss_epilogue(const v8f& acc0, const v8f& acc1,
                                              int b, int i0, int j0, int wm, int wn,
                                              int lane, int wid, int tid,
                                              const int* __restrict__ sa,
                                              const int* __restrict__ ta,
                                              const unsigned* __restrict__ bits,
                                              float* __restrict__ redN, float* __restrict__ redD,
                                              float* __restrict__ acc) {
    float numer = 0.f, denom = 0.f;
    const int rowHi = (lane >> 4) * 8;          // C/D layout: lanes 16-31 hold M = r+8
    #pragma unroll
    for (int ct = 0; ct < 2; ++ct) {
        v8f av = ct ? acc1 : acc0;
        int gj  = j0 + wn + ct * 16 + (lane & 15);
        int tav = ta[b * S_LEN + gj];
        #pragma unroll
        for (int r = 0; r < 8; ++r) {
            int gi = i0 + wm + rowHi + r;
            if (sa[b * S_LEN + gi] && tav) {
                float x = __logf(fabsf(av[r]) + EPSI);
                unsigned w = bits[((size_t)(b * S_LEN + gi) << 5) + (gj >> 5)];
                float t = ((w >> (gj & 31)) & 1u) ? 1.f : 0.f;
                numer += fmaxf(x, 0.f) - x * t + log1pf(__expf(-fabsf(x)));
                denom += 1.f;                   // weight == 1 (POS_W == NEG_W == 1)
            }
        }
    }
    #pragma unroll
    for (int off = 16; off > 0; off >>= 1) {
        numer += __shfl_down(numer, off, 32);
        denom += __shfl_down(denom, off, 32);
    }
    if (lane == 0) { redN[wid] = numer; redD[wid] = denom; }
    __syncthreads();
    if (tid == 0) {
        float n = 0.f, d = 0.f;
        for (int w = 0; w < 8; ++w) { n += redN[w]; d += redD[w]; }
        atomicAdd(&acc[0], n);
        atomicAdd(&acc[1], d);
    }
}

// ============================================================================
// Path A: async-staged GEMM over pre-materialized u_eff/v_eff
//   double-buffered LDS, tiles moved with GLOBAL_LOAD_ASYNC_TO_LDS_B128
// ============================================================================
#define BM 64
#define BN 64
#define BK2 32
#define LDP2 36            // padded LDS stride (floats): 16B aligned, conflict-free
#define TILE_F (64 * LDP2) // floats per staged tile

#define WAIT_ASYNC(n) asm volatile("s_wait_asynccnt " #n ::: "memory")

__device__ __forceinline__ void issue_tile_async(const float* __restrict__ gA,
                                                 const float* __restrict__ gB,
                                                 float* sA, float* sB, int tid, int k0) {
    #pragma unroll
    for (int it = 0; it < 2; ++it) {
        int q  = tid + it * 256;   // 0..511
        int r  = q >> 3;           // tile row 0..63
        int c4 = (q & 7) << 2;     // col (floats): 0..28
        unsigned long long ga = (unsigned long long)(uintptr_t)(gA + (size_t)r * D_DIM + k0 + c4);
        unsigned long long gb = (unsigned long long)(uintptr_t)(gB + (size_t)r * D_DIM + k0 + c4);
        unsigned la = (unsigned)(uintptr_t)(sA + r * LDP2 + c4);   // LDS byte address (low 32 bits)
        unsigned lb = (unsigned)(uintptr_t)(sB + r * LDP2 + c4);
        asm volatile("global_load_async_to_lds_b128 %0, %1, off" :: "v"(la), "v"(ga) : "memory");
        asm volatile("global_load_async_to_lds_b128 %0, %1, off" :: "v"(lb), "v"(gb) : "memory");
    }
}

__launch_bounds__(256, 2)
__global__ void k_gemm_async(const float* __restrict__ ueff, const float* __restrict__ veff,
                             const int* __restrict__ sa, const int* __restrict__ ta,
                             const unsigned* __restrict__ bits, float* __restrict__ acc) {
    __shared__ __align__(16) float smem[4 * TILE_F];   // [buf][A|B]
    __shared__ float redN[8], redD[8];

    const int b  = blockIdx.z;
    const int i0 = blockIdx.y * BM;
    const int j0 = blockIdx.x * BN;
    const int tid  = threadIdx.x;
    const int lane = tid & 31;
    const int wid  = tid >> 5;
    const int wm = (wid & 3) * 16;
    const int wn = (wid >> 2) * 32;

    const float* gA = ueff + (size_t)(b * S_LEN + i0) * D_DIM;
    const float* gB = veff + (size_t)(b * S_LEN + j0) * D_DIM;

    v8f acc0 = {0.f,0.f,0.f,0.f,0.f,0.f,0.f,0.f};
    v8f acc1 = {0.f,0.f,0.f,0.f,0.f,0.f,0.f,0.f};

    const int mA = wm + (lane & 15);
    const int nA = wn + (lane & 15);
    const int kh = (lane >> 4) << 1;       // fragment K half: 0 or 2

    const int NT = D_DIM / BK2;            // 16 K-tiles
    issue_tile_async(gA, gB, smem, smem + TILE_F, tid, 0);

    for (int t = 0; t < NT; ++t) {
        const int cur = t & 1;
        float* As_ = smem + (cur ? 2 * TILE_F : 0);
        float* Bs_ = As_ + TILE_F;
        if (t + 1 < NT) {
            float* AsN = smem + ((cur ^ 1) ? 2 * TILE_F : 0);
            issue_tile_async(gA, gB, AsN, AsN + TILE_F, tid, (t + 1) * BK2);
            WAIT_ASYNC(4);                 // tile t complete (in-order), tile t+1 in flight
        } else {
            WAIT_ASYNC(0);
        }
        __syncthreads();

        #pragma unroll
        for (int kk = 0; kk < BK2; kk += 4) {
            v2f a  = *(const v2f*)&As_[mA * LDP2 + kk + kh];
            v2f b0 = *(const v2f*)&Bs_[nA * LDP2 + kk + kh];
            v2f b1 = *(const v2f*)&Bs_[(nA + 16) * LDP2 + kk + kh];
            acc0 = __builtin_amdgcn_wmma_f32_16x16x4_f32(false, a, false, b0, (short)0, acc0, false, false);
            acc1 = __builtin_amdgcn_wmma_f32_16x16x4_f32(false, a, false, b1, (short)0, acc1, false, false);
        }
        __syncthreads();
    }

    loss_epilogue(acc0, acc1, b, i0, j0, wm, wn, lane, wid, tid, sa, ta, bits, redN, redD, acc);
}

// ============================================================================
// Path B (fallback, ws too small): scale-on-load GEMM
// ============================================================================
#define BK 64
#define LDP 68

__launch_bounds__(256, 2)
__global__ void k_gemm_loss(const float* __restrict__ u,  const float* __restrict__ us,
                            const float* __restrict__ v,  const float* __restrict__ vs,
                            const int* __restrict__ tfm,  const int* __restrict__ sa,
                            const int* __restrict__ ta,   const unsigned* __restrict__ bits,
                            float* __restrict__ acc) {
    __shared__ __align__(16) float As[BM * LDP];
    __shared__ __align__(16) float Bs[BN * LDP];
    __shared__ float redN[8], redD[8];

    const int b  = blockIdx.z;
    const int i0 = blockIdx.y * BM;
    const int j0 = blockIdx.x * BN;
    const int tid  = threadIdx.x;
    const int lane = tid & 31;
    const int wid  = tid >> 5;
    const int wm = (wid & 3) * 16;
    const int wn = (wid >> 2) * 32;

    v8f acc0 = {0.f,0.f,0.f,0.f,0.f,0.f,0.f,0.f};
    v8f acc1 = {0.f,0.f,0.f,0.f,0.f,0.f,0.f,0.f};

    const int m  = wm + (lane & 15);
    const int kh = (lane >> 4) << 1;
    const int nA = wn + (lane & 15);

    for (int k0 = 0; k0 < D_DIM; k0 += BK) {
        #pragma unroll
        for (int it = 0; it < 4; ++it) {
            int q  = tid + it * 256;
            int r  = q >> 4;
            int c4 = (q & 15) << 2;
            size_t gAo = ((size_t)(b * S_LEN + i0 + r)) * D_DIM + (k0 + c4);
            size_t gBo = ((size_t)(b * S_LEN + j0 + r)) * D_DIM + (k0 + c4);
            if (k0 + BK < D_DIM) {
                __builtin_prefetch(u + gAo + BK, 0, 1);
                __builtin_prefetch(v + gBo + BK, 0, 1);
            }
            float4 a4 = *(const float4*)(u  + gAo);
            float4 s4 = *(const float4*)(us + gAo);
            float  tf = tfm[b * S_LEN + i0 + r] ? 1.f : 0.f;
            float4 oa; oa.x = a4.x * s4.x * tf; oa.y = a4.y * s4.y * tf;
                       oa.z = a4.z * s4.z * tf; oa.w = a4.w * s4.w * tf;
            *(float4*)&As[r * LDP + c4] = oa;
            float4 b4 = *(const float4*)(v  + gBo);
            float4 t4 = *(const float4*)(vs + gBo);
            float4 ob; ob.x = b4.x * t4.x; ob.y = b4.y * t4.y;
                       ob.z = b4.z * t4.z; ob.w = b4.w * t4.w;
            *(float4*)&Bs[r * LDP + c4] = ob;
        }
        __syncthreads();

        #pragma unroll
        for (int kk = 0; kk < BK; kk += 4) {
            v2f a  = *(const v2f*)&As[m * LDP + kk + kh];
            v2f b0 = *(const v2f*)&Bs[nA * LDP + kk + kh];
            v2f b1 = *(const v2f*)&Bs[(nA + 16) * LDP + kk + kh];
            acc0 = __builtin_amdgcn_wmma_f32_16x16x4_f32(false, a, false, b0, (short)0, acc0, false, false);
            acc1 = __builtin_amdgcn_wmma_f32_16x16x4_f32(false, a, false, b1, (short)0, acc1, false, false);
        }
        __syncthreads();
    }

    loss_epilogue(acc0, acc1, b, i0, j0, wm, wn, lane, wid, tid, sa, ta, bits, redN, redD, acc);
}

__global__ void k_final(const float* __restrict__ acc, float* __restrict__ out) {
    out[0] = acc[0] / (acc[1] + EPSI);
}

// ---------------- launch ----------------
extern "C" void kernel_launch(void* const* d_in, const int* in_sizes, int n_in,
                              void* d_out, int out_size, void* d_ws, size_t ws_size,
                              hipStream_t stream) {
    const int*           gene_tokens = (const int*)d_in[0];
    const unsigned char* btg         = (const unsigned char*)d_in[1];
    const unsigned char* btf         = (const unsigned char*)d_in[2];
    const float*         u           = (const float*)d_in[3];
    const float*         v           = (const float*)d_in[4];
    const float*         us          = (const float*)d_in[5];
    const float*         vs          = (const float*)d_in[6];
    const int*           src_ids     = (const int*)d_in[7];
    const int*           tgt_ids     = (const int*)d_in[8];

    char* ws = (char*)d_ws;
    float*    accv = (float*)(ws + OFF_ACC);
    int*      used = (int*)(ws + OFF_USED);
    int*      srcf = (int*)(ws + OFF_SRCF);
    int*      tgtf = (int*)(ws + OFF_TGTF);
    int*      tfm  = (int*)(ws + OFF_TF);
    int*      sa   = (int*)(ws + OFF_SRCA);
    int*      ta   = (int*)(ws + OFF_TGTA);
    int*      pos  = (int*)(ws + OFF_POS);
    unsigned* bits = (unsigned*)(ws + OFF_BITS);
    float*    ueff = (float*)(ws + OFF_UEFF);
    float*    veff = (float*)(ws + OFF_VEFF);

    auto blocks = [](int n) { int g = (n + 255) / 256; return g > 4096 ? 4096 : g; };

    // zero / sentinel init (every call; ws is not re-poisoned between replays)
    k_fill<<<1, 256, 0, stream>>>((int*)accv, 0, 2);
    k_fill<<<blocks(V_SIZE), 256, 0, stream>>>(used, 0, V_SIZE);
    k_fill<<<blocks(V_SIZE), 256, 0, stream>>>(srcf, 0, V_SIZE);
    k_fill<<<blocks(V_SIZE), 256, 0, stream>>>(tgtf, 0, V_SIZE);
    k_fill<<<blocks(C_B * S_LEN * (S_LEN / 32)), 256, 0, stream>>>((int*)bits, 0, C_B * S_LEN * (S_LEN / 32));
    k_fill<<<blocks(C_B * V_SIZE), 256, 0, stream>>>(pos, 0x7FFFFFFF, C_B * V_SIZE);

    k_scan<<<(C_B * S_LEN + 255) / 256, 256, 0, stream>>>(gene_tokens, btg, btf, used, tfm, pos);
    k_pairs<<<(P_PAIRS + 255) / 256, 256, 0, stream>>>(src_ids, tgt_ids, used, srcf, tgtf);
    k_assoc<<<(C_B * S_LEN + 255) / 256, 256, 0, stream>>>(gene_tokens, tfm, srcf, tgtf, sa, ta);
    k_target<<<(C_B * P_PAIRS + 255) / 256, 256, 0, stream>>>(src_ids, tgt_ids, used, pos, tfm, bits);

    dim3 grid(S_LEN / BN, S_LEN / BM, C_B);   // 16 x 16 x 16 tiles

    if (ws_size >= WS_NEED_ASYNC) {
        // materialize effective factors once, then async-staged WMMA GEMM
        int nv4 = C_B * S_LEN * D_DIM / 4;
        k_eff<<<(nv4 + 255) / 256, 256, 0, stream>>>(u, us, v, vs, tfm, ueff, veff);
        k_gemm_async<<<grid, 256, 0, stream>>>(ueff, veff, sa, ta, bits, accv);
    } else {
        k_gemm_loss<<<grid, 256, 0, stream>>>(u, us, v, vs, tfm, sa, ta, bits, accv);
    }

    k_final<<<1, 1, 0, stream>>>(accv, (float*)d_out);
}